// BatchRankingLoss_27410481283421
// MI455X (gfx1250) — compile-verified
//
#include <hip/hip_runtime.h>
#include <hip/hip_bf16.h>

typedef __attribute__((ext_vector_type(2))) float v2f;
typedef __attribute__((ext_vector_type(8))) float v8f;

#define DECOYS 256
#define GAP 1.0f
#define THRESHOLD 0.1f

// One block per group of 256 decoys. Stage o/t into LDS, then evaluate all
// 256x256 pairs as 16x16 tiles. Each tile's difference matrices are produced
// by a rank-2 WMMA update:  D[m,n] = o_m*1 + 1*(-o_n) = o_m - o_n.
__global__ __launch_bounds__(256) void ranking_loss_main(
    const float* __restrict__ o, const float* __restrict__ t,
    float* __restrict__ partial) {
  __shared__ float sO[DECOYS];
  __shared__ float sT[DECOYS];
  __shared__ float sAcc[256];

  const int g   = blockIdx.x;
  const int tid = threadIdx.x;

  // stage group into LDS (input shape (B,1) f32 is flat B)
  sO[tid] = o[g * DECOYS + tid];
  sT[tid] = t[g * DECOYS + tid];
  __syncthreads();

  const int   wave = tid >> 5;
  const int   lane = tid & 31;
  const int   l16  = lane & 15;
  const float mlo  = (lane < 16) ? 1.0f : 0.0f;  // zero K=2/K=3 half of A/B

  float acc = 0.0f;

  // 256 tiles total (16x16 grid of 16x16 tiles); 32 tiles per wave,
  // loop count uniform across the wave -> EXEC stays all-ones for WMMA.
  for (int tt = 0; tt < 32; ++tt) {
    const int tileIdx = (wave << 5) + tt;
    const int bi = tileIdx >> 4;   // row block
    const int bj = tileIdx & 15;   // col block

    const float orow = sO[(bi << 4) + l16] * mlo;
    const float trow = sT[(bi << 4) + l16] * mlo;
    const float ocol = sO[(bj << 4) + l16] * mlo;
    const float tcol = sT[(bj << 4) + l16] * mlo;

    // A (16x4, f32): K0 = o_row, K1 = 1, K2 = K3 = 0
    // B (4x16, f32): K0 = 1, K1 = -o_col, K2 = K3 = 0
    v2f Ao = { orow, mlo };
    v2f Bo = { mlo, -ocol };
    v2f At = { trow, mlo };
    v2f Bt = { mlo, -tcol };
    v8f Z  = {0.f, 0.f, 0.f, 0.f, 0.f, 0.f, 0.f, 0.f};

    // D = A*B + 0 : full 16x16 tile of pairwise differences per instruction
    v8f dO = __builtin_amdgcn_wmma_f32_16x16x4_f32(
        false, Ao, false, Bo, (short)0, Z, false, false);
    v8f dT = __builtin_amdgcn_wmma_f32_16x16x4_f32(
        false, At, false, Bt, (short)0, Z, false, false);

#pragma unroll
    for (int r = 0; r < 8; ++r) {
      const float dt = dT[r];
      const float dd = dO[r];
      const float y  = (dt < 0.0f) ? -1.0f : 1.0f;          // y_ij
      const float h  = fmaxf(0.0f, GAP + y * dd);            // hinge
      acc += (fabsf(dt) > THRESHOLD) ? h : 0.0f;             // weight mask
    }
  }

  // block reduction -> one partial per group (deterministic; no float atomics)
  sAcc[tid] = acc;
  __syncthreads();
  for (int s = 128; s > 0; s >>= 1) {
    if (tid < s) sAcc[tid] += sAcc[tid + s];
    __syncthreads();
  }
  if (tid == 0) partial[g] = sAcc[0];
}

__global__ __launch_bounds__(256) void ranking_loss_reduce(
    const float* __restrict__ partial, float* __restrict__ out, int G) {
  __shared__ float sAcc[256];
  const int tid = threadIdx.x;
  float s = 0.0f;
  for (int i = tid; i < G; i += 256) s += partial[i];
  sAcc[tid] = s;
  __syncthreads();
  for (int k = 128; k > 0; k >>= 1) {
    if (tid < k) sAcc[tid] += sAcc[tid + k];
    __syncthreads();
  }
  if (tid == 0) {
    const double N = (double)G * (double)DECOYS * (double)(DECOYS - 1);
    out[0] = (float)((double)sAcc[0] / N);
  }
}

extern "C" void kernel_launch(void* const* d_in, const int* in_sizes, int n_in,
                              void* d_out, int out_size, void* d_ws, size_t ws_size,
                              hipStream_t stream) {
  const float* o = (const float*)d_in[0];   // input, (B,1) f32
  const float* t = (const float*)d_in[1];   // gdt_ts, (B,) f32
  float* out = (float*)d_out;
  float* partial = (float*)d_ws;

  const int B = in_sizes[0];
  const int K = B / DECOYS;
  const int G = K - 1;  // reference drops the final group

  ranking_loss_main<<<G, 256, 0, stream>>>(o, t, partial);
  ranking_loss_reduce<<<1, 256, 0, stream>>>(partial, out, G);
}